// TriLinearPytorch_71296457114367
// MI455X (gfx1250) — compile-verified
//
#include <hip/hip_runtime.h>

// Trilinear 3D-LUT (33^3, 3ch) applied to (4,3,1080,1920) image, f32 in/out.
// Memory-bound: ~199 MB HBM stream (~8.5us floor @ 23.3 TB/s); LUT stays in L2/WGP$.
// Strategy: repack LUT to (z,y,x,4) float4 AoS in d_ws -> 8x global_load_b128
// gathers/pixel (all cache-resident), b128 NT streaming for img/out,
// global_prefetch_b8 for the input planes. Batch index carried in blockIdx.y
// (no per-thread integer division).

typedef float v4f __attribute__((ext_vector_type(4)));

namespace {
constexpr int LUT_D  = 33;
constexpr int NLUT   = LUT_D * LUT_D * LUT_D;   // 35937 entries
constexpr int SPLANE = 1080 * 1920;             // 2,073,600 px per plane
constexpr int NBATCH = 4;
constexpr int QPLANE = SPLANE / 4;              // 518,400 quads per plane = 2025 * 256
constexpr int PF_ELEMS = 4096;                  // prefetch distance (16 KB per plane)
}

// ---- one-shot: planar (3,D,D,D) -> interleaved (D,D,D,4) float4 ----
__global__ __launch_bounds__(256) void lut_repack_kernel(
        const float* __restrict__ lut, float* __restrict__ lut4) {
    int i = blockIdx.x * blockDim.x + threadIdx.x;
    if (i >= NLUT) return;
    v4f v;
    v.x = lut[i];
    v.y = lut[NLUT + i];
    v.z = lut[2 * NLUT + i];
    v.w = 0.0f;
    *(v4f*)(lut4 + 4 * i) = v;   // regular (RT) store: keep hot in L2 for gathers
}

__device__ __forceinline__ void axis_split(float c, int& i0, int& i1, float& f) {
    float p  = fminf(fmaxf(c * 32.0f, 0.0f), 32.0f);
    float fl = floorf(p);
    i0 = (int)fl;
    i1 = min(i0 + 1, 32);
    f  = p - fl;
}

// ---- main path: packed float4 LUT in d_ws ----
__global__ __launch_bounds__(256) void trilut_packed_kernel(
        const float* __restrict__ img, const float* __restrict__ lutp,
        float* __restrict__ out) {
    int s    = (blockIdx.x * blockDim.x + threadIdx.x) * 4;  // quad start in plane
    int base = blockIdx.y * 3 * SPLANE + s;                  // 16B-aligned

    const v4f* __restrict__ L = (const v4f*)lutp;

    // stream-ahead prefetch of the three input planes (speculative, safe OOB)
    __builtin_prefetch(img + base + PF_ELEMS, 0, 1);
    __builtin_prefetch(img + base + SPLANE + PF_ELEMS, 0, 1);
    __builtin_prefetch(img + base + 2 * SPLANE + PF_ELEMS, 0, 1);

    v4f vx = __builtin_nontemporal_load((const v4f*)(img + base));
    v4f vy = __builtin_nontemporal_load((const v4f*)(img + base + SPLANE));
    v4f vz = __builtin_nontemporal_load((const v4f*)(img + base + 2 * SPLANE));

    v4f r0, r1, r2;
#pragma unroll
    for (int k = 0; k < 4; ++k) {
        int x0, x1, y0, y1, z0, z1;
        float fx, fy, fz;
        axis_split(vx[k], x0, x1, fx);
        axis_split(vy[k], y0, y1, fy);
        axis_split(vz[k], z0, z1, fz);
        float gx = 1.0f - fx, gy = 1.0f - fy, gz = 1.0f - fz;

        int zy00 = (z0 * LUT_D + y0) * LUT_D;
        int zy01 = (z0 * LUT_D + y1) * LUT_D;
        int zy10 = (z1 * LUT_D + y0) * LUT_D;
        int zy11 = (z1 * LUT_D + y1) * LUT_D;

        v4f acc = L[zy00 + x0] * (gz * gy * gx);
        acc    += L[zy00 + x1] * (gz * gy * fx);
        acc    += L[zy01 + x0] * (gz * fy * gx);
        acc    += L[zy01 + x1] * (gz * fy * fx);
        acc    += L[zy10 + x0] * (fz * gy * gx);
        acc    += L[zy10 + x1] * (fz * gy * fx);
        acc    += L[zy11 + x0] * (fz * fy * gx);
        acc    += L[zy11 + x1] * (fz * fy * fx);

        r0[k] = acc.x;
        r1[k] = acc.y;
        r2[k] = acc.z;
    }

    __builtin_nontemporal_store(r0, (v4f*)(out + base));
    __builtin_nontemporal_store(r1, (v4f*)(out + base + SPLANE));
    __builtin_nontemporal_store(r2, (v4f*)(out + base + 2 * SPLANE));
}

// ---- fallback: gather from the planar LUT directly (no workspace needed) ----
__global__ __launch_bounds__(256) void trilut_planar_kernel(
        const float* __restrict__ img, const float* __restrict__ lut,
        float* __restrict__ out) {
    int s    = (blockIdx.x * blockDim.x + threadIdx.x) * 4;
    int base = blockIdx.y * 3 * SPLANE + s;

    v4f vx = __builtin_nontemporal_load((const v4f*)(img + base));
    v4f vy = __builtin_nontemporal_load((const v4f*)(img + base + SPLANE));
    v4f vz = __builtin_nontemporal_load((const v4f*)(img + base + 2 * SPLANE));

    v4f r0, r1, r2;
#pragma unroll
    for (int k = 0; k < 4; ++k) {
        int x0, x1, y0, y1, z0, z1;
        float fx, fy, fz;
        axis_split(vx[k], x0, x1, fx);
        axis_split(vy[k], y0, y1, fy);
        axis_split(vz[k], z0, z1, fz);
        float gx = 1.0f - fx, gy = 1.0f - fy, gz = 1.0f - fz;

        int i000 = (z0 * LUT_D + y0) * LUT_D + x0;
        int i001 = (z0 * LUT_D + y0) * LUT_D + x1;
        int i010 = (z0 * LUT_D + y1) * LUT_D + x0;
        int i011 = (z0 * LUT_D + y1) * LUT_D + x1;
        int i100 = (z1 * LUT_D + y0) * LUT_D + x0;
        int i101 = (z1 * LUT_D + y0) * LUT_D + x1;
        int i110 = (z1 * LUT_D + y1) * LUT_D + x0;
        int i111 = (z1 * LUT_D + y1) * LUT_D + x1;

        float w000 = gz * gy * gx, w001 = gz * gy * fx;
        float w010 = gz * fy * gx, w011 = gz * fy * fx;
        float w100 = fz * gy * gx, w101 = fz * gy * fx;
        float w110 = fz * fy * gx, w111 = fz * fy * fx;

#pragma unroll
        for (int c = 0; c < 3; ++c) {
            const float* Lc = lut + c * NLUT;
            float acc = Lc[i000] * w000 + Lc[i001] * w001 +
                        Lc[i010] * w010 + Lc[i011] * w011 +
                        Lc[i100] * w100 + Lc[i101] * w101 +
                        Lc[i110] * w110 + Lc[i111] * w111;
            if (c == 0) r0[k] = acc;
            else if (c == 1) r1[k] = acc;
            else r2[k] = acc;
        }
    }

    __builtin_nontemporal_store(r0, (v4f*)(out + base));
    __builtin_nontemporal_store(r1, (v4f*)(out + base + SPLANE));
    __builtin_nontemporal_store(r2, (v4f*)(out + base + 2 * SPLANE));
}

extern "C" void kernel_launch(void* const* d_in, const int* in_sizes, int n_in,
                              void* d_out, int out_size, void* d_ws, size_t ws_size,
                              hipStream_t stream) {
    const float* lut = (const float*)d_in[0];   // (3,33,33,33)
    const float* img = (const float*)d_in[1];   // (4,3,1080,1920)
    float* out = (float*)d_out;                 // (4,3,1080,1920)

    const dim3 block(256);
    const dim3 grid(QPLANE / 256, NBATCH);      // (2025, 4), exact — no tail

    const size_t packed_bytes = (size_t)NLUT * 4 * sizeof(float);  // 574,992 B
    if (ws_size >= packed_bytes && d_ws != nullptr) {
        float* lut4 = (float*)d_ws;
        lut_repack_kernel<<<(NLUT + 255) / 256, block, 0, stream>>>(lut, lut4);
        trilut_packed_kernel<<<grid, block, 0, stream>>>(img, lut4, out);
    } else {
        trilut_planar_kernel<<<grid, block, 0, stream>>>(img, lut, out);
    }
}